// TransitionModel_33792802685377
// MI455X (gfx1250) — compile-verified
//
#include <hip/hip_runtime.h>
#include <math.h>

// 4 x f32 vector that lowers to global_store_b128
typedef float v4f __attribute__((ext_vector_type(4)));

#define NSTATES 8192      // num_states (= 64*64*2)
#define NMASK   8191
#define XY      64
#define THREADS 256       // 8 waves (wave32)
#define PER_THR 32        // columns per thread (8 x float4)

__global__ __launch_bounds__(THREADS) void transition_row_kernel(
    const float* __restrict__ logits,   // [NSTATES, 7]
    float* __restrict__ out)            // [NSTATES, NSTATES]
{
    const int h   = blockIdx.x;         // state index == output row
    const int tid = threadIdx.x;

    // ---- per-row log_softmax over 7 logits ----------------------------
    // Address depends only on blockIdx -> block-uniform -> compiler emits
    // scalar (SMEM) loads broadcast to the whole wave.
    const float* __restrict__ L = logits + h * 7;
    const float x0 = L[0], x1 = L[1], x2 = L[2], x3 = L[3],
                x4 = L[4], x5 = L[5], x6 = L[6];
    const float m = fmaxf(fmaxf(fmaxf(x0, x1), fmaxf(x2, x3)),
                          fmaxf(fmaxf(x4, x5), x6));
    const float s = expf(x0 - m) + expf(x1 - m) + expf(x2 - m) +
                    expf(x3 - m) + expf(x4 - m) + expf(x5 - m) +
                    expf(x6 - m);
    const float lse = m + logf(s);

    // ---- stream-fill this row with -inf (non-temporal b128 stores) ----
    float* __restrict__ row = out + (size_t)h * NSTATES;
    const float NEG_INF = -__builtin_inff();
    const v4f fill = { NEG_INF, NEG_INF, NEG_INF, NEG_INF };
    const int base = tid * PER_THR;      // 128-byte aligned chunk
#pragma unroll
    for (int j = 0; j < PER_THR / 4; ++j) {
        __builtin_nontemporal_store(fill, (v4f*)(row + base + j * 4));
    }

    // ---- patch the 6 finite entries ----------------------------------
    // Offset +8192 == 0 (mod N): neighbor idx 6 overwrites idx 0 on the
    // diagonal (last-write-wins in the reference scatter).
    // Each patch is done by the SAME thread that filled that chunk, after
    // its fill stores; same-wave stores to the same address are ordered,
    // so no barrier/fence is required.
    int c;
    c = h;                     if ((c >> 5) == tid) row[c] = x6 - lse; // offset 0 / 8192
    c = (h - 1)    & NMASK;    if ((c >> 5) == tid) row[c] = x1 - lse; // (+1,0,0)
    c = (h + 1)    & NMASK;    if ((c >> 5) == tid) row[c] = x2 - lse; // (-1,0,0)
    c = (h - XY)   & NMASK;    if ((c >> 5) == tid) row[c] = x3 - lse; // (0,+1,0)
    c = (h + XY)   & NMASK;    if ((c >> 5) == tid) row[c] = x4 - lse; // (0,-1,0)
    c = (h - 4096) & NMASK;    if ((c >> 5) == tid) row[c] = x5 - lse; // (0,0,1)
}

extern "C" void kernel_launch(void* const* d_in, const int* in_sizes, int n_in,
                              void* d_out, int out_size, void* d_ws, size_t ws_size,
                              hipStream_t stream)
{
    const float* logits = (const float*)d_in[0];   // [8192, 7] fp32
    float* out = (float*)d_out;                    // [8192, 8192] fp32

    const int n_rows = in_sizes[0] / 7;            // 8192
    transition_row_kernel<<<n_rows, THREADS, 0, stream>>>(logits, out);
}